// SMLossAVT_24172075941904
// MI455X (gfx1250) — compile-verified
//
#include <hip/hip_runtime.h>
#include <stdint.h>

// ---------------------------------------------------------------------------
// SMLossAVT on MI455X (gfx1250)
//   - TDM (tensor_load_to_lds) streams 16x128 f32 chunks into padded LDS,
//     double-buffered on TENSORcnt.
//   - V_WMMA_F32_16X16X4_F32 with an all-ones B matrix computes 16 row-sums
//     per instruction (sum and sum-of-squares accumulators).
//   - Symmetric-KL log terms cancel exactly; final scalar = sum(g)/384.
// ---------------------------------------------------------------------------

typedef float        v2f __attribute__((ext_vector_type(2)));
typedef float        v8f __attribute__((ext_vector_type(8)));
typedef unsigned int v4u __attribute__((ext_vector_type(4)));
typedef int          v4i __attribute__((ext_vector_type(4)));
typedef int          v8i __attribute__((ext_vector_type(8)));

#define D_FEAT      1024
#define ROWS_TOT    16384                  // 32*512 rows per tensor
#define TILE_ROWS   16
#define TILES_PER_T (ROWS_TOT / TILE_ROWS) // 1024
#define CHUNK_COLS  128
#define NCHUNK      (D_FEAT / CHUNK_COLS)  // 8
#define LDS_STRIDE  132                    // 128 + 4 pad floats -> conflict-free b64
#define CHUNK_LDS_FLOATS (TILE_ROWS * LDS_STRIDE) // 2112 floats = 8448 B

// One TDM descriptor: 2D tile, 16 rows x 128 f32, tensor row stride 1024 f32,
// LDS padding of 4 DWORDs after every 128 DWORDs (row stride 132 floats).
__device__ __forceinline__ void tdm_load_chunk(uint64_t gaddr, uint32_t lds_byte_addr)
{
    v4u g0;
    g0.x = 1u;                                             // count=1, user descriptor
    g0.y = lds_byte_addr;                                  // lds_addr [63:32]
    g0.z = (uint32_t)(gaddr & 0xFFFFFFFFull);              // global_addr [95:64]
    g0.w = (uint32_t)((gaddr >> 32) & 0x01FFFFFFull)       // global_addr [120:96]
         | (2u << 30);                                     // type = 2 ("image")

    v8i g1 = {
        (int)((2u << 16)      // data_size = 4 bytes
            | (1u << 20)      // pad_enable
            | (6u << 22)      // pad_interval: 2<<6 = 128 DWORDs
            | (3u << 25)),    // pad_amount: 3+1 = 4 DWORDs
        (int)(128u << 16),    // tensor_dim0[15:0] = 128 (chunk width)
        (int)(16u  << 16),    // tensor_dim1[15:0] = 16
        (int)(128u << 16),    // tile_dim0 = 128
        16,                   // tile_dim1 = 16
        1024,                 // tensor_dim0_stride[31:0] = 1024 f32 (full row)
        0, 0
    };
    v4i z4 = {0, 0, 0, 0};        // 2D tensor: groups 2/3 unused
    v8i z8 = {0, 0, 0, 0, 0, 0, 0, 0};
    __builtin_amdgcn_tensor_load_to_lds(g0, g1, z4, z4, z8, 0);
}

// One wave per 16x1024 tile: TDM double-buffered load, WMMA row reduction.
__global__ __launch_bounds__(32)
void smloss_stats_kernel(const float* __restrict__ x0,
                         const float* __restrict__ x1,
                         const float* __restrict__ x2,
                         float* __restrict__ mu,
                         float* __restrict__ var)
{
    __shared__ float lbuf[2][CHUNK_LDS_FLOATS];

    const int task = blockIdx.x;            // 0 .. 3*1024-1
    const int t    = task >> 10;            // tensor index
    const int tile = task & 1023;           // tile within tensor
    const float* src = (t == 0) ? x0 : (t == 1) ? x1 : x2;

    const uint64_t base = (uint64_t)(uintptr_t)src
                        + (uint64_t)tile * (TILE_ROWS * D_FEAT * 4u);

    // Flat->LDS: low 32 bits of a shared-memory pointer are the LDS byte addr.
    const uint32_t lds0 = (uint32_t)(uintptr_t)&lbuf[0][0];
    const uint32_t lds1 = (uint32_t)(uintptr_t)&lbuf[1][0];

    const int lane  = threadIdx.x;
    const int row   = lane & 15;            // A-matrix M = lane % 16
    const int khalf = (lane >> 4) << 1;     // lanes 16-31 hold K+2,K+3

    v8f csum = {};                          // row sums (broadcast over N)
    v8f csq  = {};                          // row sums of squares
    const v2f ones = {1.0f, 1.0f};          // B = ones(4x16): layout-invariant

    tdm_load_chunk(base, lds0);
    for (int c = 0; c < NCHUNK; ++c) {
        if (c + 1 < NCHUNK) {
            tdm_load_chunk(base + (uint64_t)(c + 1) * (CHUNK_COLS * 4u),
                           ((c + 1) & 1) ? lds1 : lds0);
            __builtin_amdgcn_s_wait_tensorcnt(1);   // chunk c landed (in-order)
        } else {
            __builtin_amdgcn_s_wait_tensorcnt(0);
        }

        const float* bp = &lbuf[c & 1][row * LDS_STRIDE + khalf];
#pragma unroll
        for (int j = 0; j < CHUNK_COLS / 4; ++j) {
            v2f a  = *(const v2f*)(bp + 4 * j);     // ds_load_b64, conflict-free
            v2f a2 = a * a;
            csum = __builtin_amdgcn_wmma_f32_16x16x4_f32(
                       false, a,  false, ones, (short)0, csum, false, false);
            csq  = __builtin_amdgcn_wmma_f32_16x16x4_f32(
                       false, a2, false, ones, (short)0, csq,  false, false);
        }
    }

    // D layout: VGPR j holds row j (lanes 0-15) and row j+8 (lanes 16-31),
    // value replicated across N. Lanes 0 and 16 publish the 16 rows.
    if (lane == 0 || lane == 16) {
        const int rbase = t * ROWS_TOT + tile * TILE_ROWS + ((lane >> 4) << 3);
#pragma unroll
        for (int j = 0; j < 8; ++j) {
            const float s = csum[j];
            const float q = csq[j];
            mu[rbase + j]  = s * (1.0f / 1024.0f);
            var[rbase + j] = (q - s * s * (1.0f / 1024.0f)) * (1.0f / 1023.0f);
        }
    }
}

// Per-row symmetric-KL integrand, block-reduced to 64 partials.
__global__ __launch_bounds__(256)
void smloss_kl_partial_kernel(const float* __restrict__ mu,
                              const float* __restrict__ var,
                              float* __restrict__ partial)
{
    __shared__ float red[256];
    const int i = blockIdx.x * 256 + threadIdx.x;

    const float m0 = mu[0 * ROWS_TOT + i];
    const float m1 = mu[1 * ROWS_TOT + i];
    const float m2 = mu[2 * ROWS_TOT + i];
    const float v0 = var[0 * ROWS_TOT + i];
    const float v1 = var[1 * ROWS_TOT + i];
    const float v2 = var[2 * ROWS_TOT + i];
    const float i0 = 1.0f / v0, i1 = 1.0f / v1, i2 = 1.0f / v2;

    float g = 0.0f;
    { const float d = m0 - m1; g += v0 * i1 + v1 * i0 + d * d * (i0 + i1) - 2.0f; }
    { const float d = m0 - m2; g += v0 * i2 + v2 * i0 + d * d * (i0 + i2) - 2.0f; }
    { const float d = m1 - m2; g += v1 * i2 + v2 * i1 + d * d * (i1 + i2) - 2.0f; }

    red[threadIdx.x] = g;
    __syncthreads();
    for (int s = 128; s > 0; s >>= 1) {
        if (threadIdx.x < s) red[threadIdx.x] += red[threadIdx.x + s];
        __syncthreads();
    }
    if (threadIdx.x == 0) partial[blockIdx.x] = red[0];
}

__global__ __launch_bounds__(64)
void smloss_final_kernel(const float* __restrict__ partial, float* __restrict__ out)
{
    __shared__ float red[64];
    red[threadIdx.x] = partial[threadIdx.x];
    __syncthreads();
    for (int s = 32; s > 0; s >>= 1) {
        if (threadIdx.x < s) red[threadIdx.x] += red[threadIdx.x + s];
        __syncthreads();
    }
    // out = (1/3) * sum over 3 pairs of (KLxy+KLyx)/2, batch-mean 1/32, 0.5*0.5
    //     = sum_rows(g) / 384
    if (threadIdx.x == 0) out[0] = red[0] * (1.0f / 384.0f);
}

extern "C" void kernel_launch(void* const* d_in, const int* in_sizes, int n_in,
                              void* d_out, int out_size, void* d_ws, size_t ws_size,
                              hipStream_t stream)
{
    const float* V = (const float*)d_in[0];
    const float* A = (const float*)d_in[1];
    const float* T = (const float*)d_in[2];

    float* mu      = (float*)d_ws;            // 3 * 16384 floats
    float* var     = mu  + 3 * ROWS_TOT;      // 3 * 16384 floats
    float* partial = var + 3 * ROWS_TOT;      // 64 floats

    smloss_stats_kernel<<<3 * TILES_PER_T, 32, 0, stream>>>(V, A, T, mu, var);
    smloss_kl_partial_kernel<<<ROWS_TOT / 256, 256, 0, stream>>>(mu, var, partial);
    smloss_final_kernel<<<1, 64, 0, stream>>>(partial, (float*)d_out);
}